// Temporal_Attention_layer_61065845015282
// MI455X (gfx1250) — compile-verified
//
#include <hip/hip_runtime.h>
#include <math.h>

typedef __attribute__((ext_vector_type(2))) float v2f;
typedef __attribute__((ext_vector_type(8))) float v8f;
typedef unsigned int u32x4 __attribute__((ext_vector_type(4)));
typedef int i32x4 __attribute__((ext_vector_type(4)));
typedef int i32x8 __attribute__((ext_vector_type(8)));

#define B_ 32
#define F_ 64
#define N_ 512
#define T_ 64

// ---------------------------------------------------------------- zero init
__global__ void zero_f32(float* p, int n) {
    int i = blockIdx.x * blockDim.x + threadIdx.x;
    if (i < n) p[i] = 0.f;
}

// ---------------------------------------------------------------- TDM helper:
// async 2D tile copy (64 x 64 x fp32, row-major, contiguous) global -> LDS,
// descriptor per CDNA5 ISA ch.8 (D# group0/group1), groups 2/3 zero (2D tile).
// This toolchain exposes the 6-arg builtin: (g0, g1, g2, g3, g4, cpol).
__device__ __forceinline__ void tdm_load_A64x64(const float* gsrc, void* lds_dst) {
    unsigned lds_addr = (unsigned)(uintptr_t)lds_dst;     // LDS offset = addr[31:0]
    unsigned long long ga = (unsigned long long)(uintptr_t)gsrc;

    u32x4 g0;
    g0.x = 0x1u;                                   // count=1 (valid user D#)
    g0.y = lds_addr;                               // lds_addr[31:0]
    g0.z = (unsigned)(ga & 0xFFFFFFFFu);           // global_addr[31:0]
    g0.w = (unsigned)(ga >> 32) | 0x80000000u;     // global_addr[56:32] | type=2<<30

    i32x8 g1;
    g1[0] = 0x00020000;            // wg_mask=0, data_size=2 (4B), no pad/iterate
    g1[1] = (int)(64u << 16);      // [31:16] = tensor_dim0[15:0] = 64
    g1[2] = (int)(64u << 16);      // [15:0]=tensor_dim0[31:16]=0, [31:16]=tensor_dim1[15:0]=64
    g1[3] = (int)(64u << 16);      // [15:0]=tensor_dim1[31:16]=0, [31:16]=tile_dim0=64
    g1[4] = 64;                    // [15:0]=tile_dim1=64, [31:16]=tile_dim2=0
    g1[5] = 64;                    // tensor_dim0_stride[31:0] = 64 elements
    g1[6] = (int)(4096u << 16);    // dim0_stride[47:32]=0, dim1_stride[15:0]=4096
    g1[7] = 0;                     // dim1_stride[47:16]=0

    i32x4 z4 = {0, 0, 0, 0};
    i32x8 z8 = {0, 0, 0, 0, 0, 0, 0, 0};
    __builtin_amdgcn_tensor_load_to_lds(g0, g1, z4, z4, z8, 0);
}

// ---------------------------------------------------------------- pass 1:
// one streaming read of x computes BOTH
//   t1[b,f,t]  = sum_n U1[n] * x[b,f,n,t]   (via LDS reduce + global atomics)
//   rhs[b,n,t] = sum_f U3[f] * x[b,f,n,t]   (register accumulators)
__global__ void __launch_bounds__(256)
fused_t1_rhs(const float* __restrict__ x, const float* __restrict__ U1,
             const float* __restrict__ U3, float* __restrict__ t1,
             float* __restrict__ rhs) {
    int b  = blockIdx.y;
    int n0 = blockIdx.x * 32;          // 16 chunks of 32 n-values
    int tid = threadIdx.x;
    int t  = tid & 63;
    int ng = tid >> 6;                 // 0..3

    const float* xb = x + (size_t)b * F_ * N_ * T_;
    float racc[8];
#pragma unroll
    for (int k = 0; k < 8; ++k) racc[k] = 0.f;

    __shared__ float red[4][64];

    for (int f = 0; f < F_; ++f) {
        float u3 = U3[f];
        const float* xf = xb + (size_t)f * N_ * T_;
        float ts = 0.f;
#pragma unroll
        for (int k = 0; k < 8; ++k) {
            int n = n0 + ng * 8 + k;
            float v = xf[n * T_ + t];
            racc[k] += u3 * v;
            ts += U1[n] * v;
        }
        red[ng][t] = ts;
        __syncthreads();
        if (ng == 0) {
            float tot = red[0][t] + red[1][t] + red[2][t] + red[3][t];
            atomicAdd(&t1[(b * F_ + f) * T_ + t], tot);
        }
        __syncthreads();
    }
#pragma unroll
    for (int k = 0; k < 8; ++k) {
        int n = n0 + ng * 8 + k;
        rhs[((size_t)b * N_ + n) * T_ + t] = racc[k];
    }
}

// ---------------------------------------------------------------- pass 2:
// lhs[b,n,t] = sum_f U2[n,f] * t1[b,f,t]    (tiny, VALU)
__global__ void __launch_bounds__(256)
lhs_kernel(const float* __restrict__ U2, const float* __restrict__ t1,
           float* __restrict__ lhs) {
    int gid = blockIdx.x * blockDim.x + threadIdx.x;  // B*N*T threads
    int b   = gid >> 15;          // N_*T_ = 32768
    int rem = gid & 32767;
    int n   = rem >> 6;
    int t   = rem & 63;
    const float* t1b = t1 + b * F_ * T_;
    const float* u2n = U2 + n * F_;
    float acc = 0.f;
#pragma unroll 8
    for (int f = 0; f < F_; ++f) acc += u2n[f] * t1b[f * T_ + t];
    lhs[((size_t)b * N_ + n) * T_ + t] = acc;
}

// ---------------------------------------------------------------- pass 3:
// per-batch: product = lhs^T @ rhs (K=512, WMMA f32), sigmoid(+be),
// E = Ve @ S (K=64, WMMA f32), softmax over axis t -> Aout[b,t,u]
// 512 threads = 16 waves; wave (wi,wj) owns a 16x16 tile.
__global__ void __launch_bounds__(512)
attn_kernel(const float* __restrict__ lhs, const float* __restrict__ rhs,
            const float* __restrict__ be, const float* __restrict__ Ve,
            float* __restrict__ Aout) {
    int b    = blockIdx.x;
    int tid  = threadIdx.x;
    int lane = tid & 31;
    int w    = tid >> 5;     // 0..15
    int wi   = w >> 2;       // t-tile 0..3
    int wj   = w & 3;        // s-tile 0..3
    int l16  = lane & 15;
    int hi   = lane >> 4;    // lanes 16-31 hold K+2/K+3

    const float* lb = lhs + (size_t)b * N_ * T_;
    const float* rb = rhs + (size_t)b * N_ * T_;

    int trow = 16 * wi + l16;
    int scol = 16 * wj + l16;

    v8f c = {};
    for (int k = 0; k < N_; k += 4) {
        int ka = k + 2 * hi;
        v2f a, bb;
        a.x  = lb[ka * T_ + trow];         // A[m=t][k]   = lhs[k][t]
        a.y  = lb[(ka + 1) * T_ + trow];
        bb.x = rb[ka * T_ + scol];         // B[k][n=s]   = rhs[k][s]
        bb.y = rb[(ka + 1) * T_ + scol];
        c = __builtin_amdgcn_wmma_f32_16x16x4_f32(false, a, false, bb,
                                                  (short)0, c, false, false);
    }

    __shared__ float S[T_ * T_];
    __shared__ float Es[T_ * T_];
#pragma unroll
    for (int r = 0; r < 8; ++r) {
        int t = 16 * wi + r + 8 * hi;      // C layout: M = r + 8*(lane>=16)
        int s = 16 * wj + l16;
        float v = c[r] + be[t * T_ + s];
        S[t * T_ + s] = 1.f / (1.f + __expf(-v));
    }
    __syncthreads();

    v8f e = {};
#pragma unroll
    for (int k = 0; k < T_; k += 4) {
        int ka = k + 2 * hi;
        v2f a = *(const v2f*)(Ve + trow * T_ + ka);  // Ve row-major, aligned pair
        v2f bb;
        bb.x = S[ka * T_ + scol];
        bb.y = S[(ka + 1) * T_ + scol];
        e = __builtin_amdgcn_wmma_f32_16x16x4_f32(false, a, false, bb,
                                                  (short)0, e, false, false);
    }
#pragma unroll
    for (int r = 0; r < 8; ++r) {
        int t = 16 * wi + r + 8 * hi;
        int u = 16 * wj + l16;
        Es[t * T_ + u] = e[r];
    }
    __syncthreads();

    // softmax over t (axis=1) per column u
    if (tid < T_) {
        int u = tid;
        float m = -1e30f;
        for (int t = 0; t < T_; ++t) m = fmaxf(m, Es[t * T_ + u]);
        float sum = 0.f;
        for (int t = 0; t < T_; ++t) sum += __expf(Es[t * T_ + u] - m);
        float inv = 1.f / sum;
        for (int t = 0; t < T_; ++t)
            Aout[((size_t)b * T_ + t) * T_ + u] = __expf(Es[t * T_ + u] - m) * inv;
    }
}

// ---------------------------------------------------------------- pass 4:
// out[b, row, s] = sum_t x[b, row, t] * A[b, t, s],  row = (f,n) in [0, F*N)
// 256 threads = 8 waves = 2(m) x 4(s) tiles; block tile = 32 rows x 64 s.
// A[b] (16 KB) staged into LDS by the Tensor Data Mover (wave 0 issues,
// TENSORcnt wait, then block barrier). x A-fragments are aligned v2f loads.
__global__ void __launch_bounds__(256)
out_kernel(const float* __restrict__ x, const float* __restrict__ Aattn,
           float* __restrict__ out) {
    int b   = blockIdx.y;
    int m0  = blockIdx.x * 32;
    int tid = threadIdx.x;
    int lane = tid & 31;
    int w    = tid >> 5;    // 0..7
    int wm   = w >> 2;      // 0..1
    int wsj  = w & 3;       // 0..3
    int l16  = lane & 15;
    int hi   = lane >> 4;

    __shared__ float As[T_ * T_];
    if (tid < 32) {                       // wave 0 drives the TDM (EXEC ignored)
        tdm_load_A64x64(Aattn + (size_t)b * T_ * T_, (void*)As);
        __builtin_amdgcn_s_wait_tensorcnt(0);
    }
    __syncthreads();

    const float* xb = x + (size_t)b * F_ * N_ * T_;
    int mrow = m0 + 16 * wm + l16;
    int s0   = 16 * wsj;

    v8f c = {};
#pragma unroll
    for (int k = 0; k < T_; k += 4) {
        int ka = k + 2 * hi;
        v2f a = *(const v2f*)(xb + (size_t)mrow * T_ + ka);  // contiguous t-pair
        v2f bb;
        bb.x = As[ka * T_ + s0 + l16];
        bb.y = As[(ka + 1) * T_ + s0 + l16];
        c = __builtin_amdgcn_wmma_f32_16x16x4_f32(false, a, false, bb,
                                                  (short)0, c, false, false);
    }

    float* ob = out + (size_t)b * F_ * N_ * T_;
#pragma unroll
    for (int r = 0; r < 8; ++r) {
        int row = m0 + 16 * wm + r + 8 * hi;
        ob[(size_t)row * T_ + s0 + l16] = c[r];
    }
}

// ----------------------------------------------------------------
extern "C" void kernel_launch(void* const* d_in, const int* in_sizes, int n_in,
                              void* d_out, int out_size, void* d_ws, size_t ws_size,
                              hipStream_t stream) {
    const float* x  = (const float*)d_in[0];
    const float* U1 = (const float*)d_in[1];
    const float* U2 = (const float*)d_in[2];
    const float* U3 = (const float*)d_in[3];
    const float* be = (const float*)d_in[4];
    const float* Ve = (const float*)d_in[5];
    float* out = (float*)d_out;

    float* ws    = (float*)d_ws;
    float* t1    = ws;                  // B*F*T   = 131072 floats
    float* rhs   = t1 + 131072;         // B*N*T   = 1048576
    float* lhs   = rhs + 1048576;       // B*N*T   = 1048576
    float* Aattn = lhs + 1048576;       // B*T*T   = 131072
    (void)in_sizes; (void)n_in; (void)out_size; (void)ws_size;

    zero_f32<<<131072 / 256, 256, 0, stream>>>(t1, 131072);
    fused_t1_rhs<<<dim3(N_ / 32, B_), 256, 0, stream>>>(x, U1, U3, t1, rhs);
    lhs_kernel<<<(B_ * N_ * T_) / 256, 256, 0, stream>>>(U2, t1, lhs);
    attn_kernel<<<B_, 512, 0, stream>>>(lhs, rhs, be, Ve, Aattn);
    out_kernel<<<dim3(F_ * N_ / 32, B_), 256, 0, stream>>>(x, Aattn, out);
}